// ChunkFlowClassifier_15272903704875
// MI455X (gfx1250) — compile-verified
//
#include <hip/hip_runtime.h>
#include <hip/hip_bf16.h>
#include <math.h>

#define BATCH 64
#define SEQ 2048
#define HID 768
#define NCHUNK 16
#define CHUNK_S (SEQ / NCHUNK)   // 128
#define H4 (HID / 4)             // 192 float4 per row

typedef __attribute__((ext_vector_type(2))) float v2f;
typedef __attribute__((ext_vector_type(8))) float v8f;

// ---------------------------------------------------------------------------
// 1) valid length per batch (mask is a contiguous prefix of ones)
// ---------------------------------------------------------------------------
__global__ void len_kernel(const int* __restrict__ mask, int* __restrict__ Larr) {
    int b = blockIdx.x;
    int tid = threadIdx.x;
    int s = 0;
    for (int i = tid; i < SEQ; i += 256) s += mask[b * SEQ + i];
    __shared__ int red[256];
    red[tid] = s;
    __syncthreads();
    for (int off = 128; off > 0; off >>= 1) {
        if (tid < off) red[tid] += red[tid + off];
        __syncthreads();
    }
    if (tid == 0) Larr[b] = red[0];
}

// ---------------------------------------------------------------------------
// 2) streaming masked pooling: deterministic two-stage reduction.
//    grid = (NCHUNK, BATCH), block = 192 threads, each owns 4 consecutive h.
//    Skips s >= L-1 entirely (saves ~half the HBM traffic on average).
// ---------------------------------------------------------------------------
__global__ void pool_kernel(const float* __restrict__ hidden,
                            const int* __restrict__ Larr,
                            float* __restrict__ partials) {
    int b = blockIdx.y, chunk = blockIdx.x;
    int L = Larr[b];
    int mid = L >> 1;
    int endp = L - 1;
    int s_lo = chunk * CHUNK_S;
    int s_hi = s_lo + CHUNK_S;
    int s0 = (s_lo < 1) ? 1 : s_lo;
    int s1 = (s_hi < endp) ? s_hi : endp;

    float4 aF = make_float4(0.f, 0.f, 0.f, 0.f);
    float4 aS = make_float4(0.f, 0.f, 0.f, 0.f);
    const float4* hp = (const float4*)(hidden + (size_t)b * SEQ * HID);
    int h4 = threadIdx.x;  // 0..191

    for (int s = s0; s < s1; ++s) {
        float4 v = hp[(size_t)s * H4 + h4];
        if (s < mid) {
            aF.x += v.x; aF.y += v.y; aF.z += v.z; aF.w += v.w;
        } else {
            aS.x += v.x; aS.y += v.y; aS.z += v.z; aS.w += v.w;
        }
    }
    float4* out = (float4*)(partials + ((size_t)(b * NCHUNK + chunk) * 2) * HID);
    out[h4] = aF;         // first-half partial
    out[H4 + h4] = aS;    // second-half partial
}

// ---------------------------------------------------------------------------
// 3) reduce chunks, mean-divide, two-pass LayerNorm, build flow [B, 3H]
//    block = 256 threads, each owns 3 h positions.
// ---------------------------------------------------------------------------
__global__ void finalize_pool(const float* __restrict__ partials,
                              const int* __restrict__ Larr,
                              const float* __restrict__ gamma,
                              const float* __restrict__ beta,
                              float* __restrict__ flow) {
    int b = blockIdx.x;
    int tid = threadIdx.x;
    int L = Larr[b];
    int mid = L >> 1;
    float cF = (float)(mid - 1);       if (cF < 1e-9f) cF = 1e-9f;
    float cS = (float)(L - 1 - mid);   if (cS < 1e-9f) cS = 1e-9f;

    float fh[3], sh[3];
    for (int j = 0; j < 3; ++j) {
        int h = tid + j * 256;
        float sf = 0.f, ss = 0.f;
        for (int c = 0; c < NCHUNK; ++c) {
            const float* p = partials + ((size_t)(b * NCHUNK + c) * 2) * HID;
            sf += p[h];
            ss += p[HID + h];
        }
        fh[j] = sf / cF;
        sh[j] = ss / cS;
    }

    __shared__ float red[256];
    auto block_sum = [&](float v) -> float {
        red[tid] = v;
        __syncthreads();
        for (int off = 128; off > 0; off >>= 1) {
            if (tid < off) red[tid] += red[tid + off];
            __syncthreads();
        }
        float r = red[0];
        __syncthreads();
        return r;
    };

    const float invH = 1.0f / (float)HID;
    float muF = block_sum(fh[0] + fh[1] + fh[2]) * invH;
    float muS = block_sum(sh[0] + sh[1] + sh[2]) * invH;
    float dF = 0.f, dS = 0.f;
    for (int j = 0; j < 3; ++j) {
        float a = fh[j] - muF; dF += a * a;
        float c = sh[j] - muS; dS += c * c;
    }
    float vF = block_sum(dF) * invH;
    float vS = block_sum(dS) * invH;
    float rsF = rsqrtf(vF + 1e-5f);
    float rsS = rsqrtf(vS + 1e-5f);

    float* frow = flow + (size_t)b * (3 * HID);
    for (int j = 0; j < 3; ++j) {
        int h = tid + j * 256;
        float g = gamma[h], be = beta[h];
        float fn = (fh[j] - muF) * rsF * g + be;
        float sn = (sh[j] - muS) * rsS * g + be;
        frow[h] = fn;
        frow[HID + h] = sn;
        frow[2 * HID + h] = sn - fn;
    }
}

// ---------------------------------------------------------------------------
// 4) fp32 WMMA GEMM: C[M,N] = act(A[M,K] @ B[K,N] + bias), one 16x16 tile per
//    wave, K stepped by 4 via V_WMMA_F32_16X16X4_F32. Wave counts are exact
//    multiples of tiles -> EXEC all-1s for every WMMA.
//    A-frag layout (ISA 7.12.2): lanes 0-15 hold K=k,k+1; lanes 16-31 K=k+2,k+3.
//    B-frag: lane = N column (idx), rows striped by lane-half like A.
// ---------------------------------------------------------------------------
__device__ __forceinline__ float gelu_exact(float x) {
    return 0.5f * x * (1.0f + erff(x * 0.7071067811865475f));
}

__global__ void wmma_gemm_kernel(const float* __restrict__ A,
                                 const float* __restrict__ Bm,
                                 const float* __restrict__ bias,
                                 float* __restrict__ C,
                                 int M, int N, int K, int apply_gelu) {
    int wave = (int)((blockIdx.x * blockDim.x + threadIdx.x) >> 5);
    int lane = threadIdx.x & 31;
    int tiles_m = M >> 4, tiles_n = N >> 4;
    if (wave >= tiles_m * tiles_n) return;   // wave-uniform exit
    int tm = wave % tiles_m;
    int tn = wave / tiles_m;
    int half = lane >> 4;      // 0: K pair {0,1}, 1: K pair {2,3}
    int idx = lane & 15;       // M row (A) / N col (B,C,D)

    int row_a = tm * 16 + idx;
    int col_b = tn * 16 + idx;
    const float2* arow = (const float2*)(A + (size_t)row_a * K);  // K even, 8B aligned

    v8f c = {};
#pragma unroll 4
    for (int k = 0; k < K; k += 4) {
        float2 af = arow[(k >> 1) + half];
        v2f a; a.x = af.x; a.y = af.y;
        v2f bf;
        bf.x = Bm[(size_t)(k + 2 * half) * N + col_b];
        bf.y = Bm[(size_t)(k + 2 * half + 1) * N + col_b];
        c = __builtin_amdgcn_wmma_f32_16x16x4_f32(
            /*neg_a=*/false, a, /*neg_b=*/false, bf,
            /*c_mod=*/(short)0, c, /*reuse_a=*/false, /*reuse_b=*/false);
    }

#pragma unroll
    for (int r = 0; r < 8; ++r) {
        int row = tm * 16 + r + half * 8;   // C/D layout: lanes 16-31 -> M+8
        int col = tn * 16 + idx;
        float v = c[r] + bias[col];
        if (apply_gelu) v = gelu_exact(v);
        C[(size_t)row * N + col] = v;
    }
}

// ---------------------------------------------------------------------------
// 5) tiny head: [64,128] @ [128,5] + b3 -> [64,5]
// ---------------------------------------------------------------------------
__global__ void head_kernel(const float* __restrict__ x2,
                            const float* __restrict__ W3,
                            const float* __restrict__ b3,
                            float* __restrict__ out) {
    int o = blockIdx.x * blockDim.x + threadIdx.x;
    if (o >= BATCH * 5) return;
    int b = o / 5, j = o % 5;
    float acc = b3[j];
#pragma unroll 4
    for (int k = 0; k < 128; ++k) acc += x2[b * 128 + k] * W3[k * 5 + j];
    out[o] = acc;
}

// ---------------------------------------------------------------------------
extern "C" void kernel_launch(void* const* d_in, const int* in_sizes, int n_in,
                              void* d_out, int out_size, void* d_ws, size_t ws_size,
                              hipStream_t stream) {
    const float* hidden = (const float*)d_in[0];
    const int*   mask   = (const int*)d_in[1];
    const float* gamma  = (const float*)d_in[2];
    const float* beta   = (const float*)d_in[3];
    const float* W1     = (const float*)d_in[4];
    const float* b1     = (const float*)d_in[5];
    const float* W2     = (const float*)d_in[6];
    const float* b2     = (const float*)d_in[7];
    const float* W3     = (const float*)d_in[8];
    const float* b3     = (const float*)d_in[9];
    float* out = (float*)d_out;

    char* ws = (char*)d_ws;
    int*   Larr     = (int*)ws;                                   // 64 ints
    float* partials = (float*)(ws + 256);                         // B*NCHUNK*2*H
    size_t pelems   = (size_t)BATCH * NCHUNK * 2 * HID;
    float* flow     = partials + pelems;                          // 64 x 2304
    float* x1       = flow + (size_t)BATCH * 3 * HID;             // 64 x 512
    float* x2       = x1 + (size_t)BATCH * 512;                   // 64 x 128

    len_kernel<<<BATCH, 256, 0, stream>>>(mask, Larr);
    pool_kernel<<<dim3(NCHUNK, BATCH), 192, 0, stream>>>(hidden, Larr, partials);
    finalize_pool<<<BATCH, 256, 0, stream>>>(partials, Larr, gamma, beta, flow);

    // GEMM1: [64,2304]@[2304,512] -> 4*32=128 tiles -> 128 waves = 16 blocks x 8 waves
    wmma_gemm_kernel<<<16, 256, 0, stream>>>(flow, W1, b1, x1, BATCH, 512, 3 * HID, 1);
    // GEMM2: [64,512]@[512,128] -> 4*8=32 tiles -> 32 waves = 4 blocks x 8 waves
    wmma_gemm_kernel<<<4, 256, 0, stream>>>(x1, W2, b2, x2, BATCH, 128, 512, 1);

    head_kernel<<<1, 320, 0, stream>>>(x2, W3, b3, out);
}